// FourierAttention_42537356099959
// MI455X (gfx1250) — compile-verified
//
#include <hip/hip_runtime.h>

// Fourier attention for MI455X (gfx1250, wave32, WMMA).
// B=4, N=256, C=512, H=8, Dh=64.

typedef float v2f __attribute__((ext_vector_type(2)));
typedef float v8f __attribute__((ext_vector_type(8)));

#define WMMA_F32(a, b, c) \
    __builtin_amdgcn_wmma_f32_16x16x4_f32(false, (a), false, (b), (short)0, (c), false, false)

// ---------------------------------------------------------------------------
// Kernel 1: qkv = x(1024x512) @ w_qkv^T(512x1536), scattered into per-head
// (B,H,N,Dh) q/k/v buffers. 8 waves/block, each wave owns a 16x16 tile.
// Block tile = 32(M) x 64(N). Grid = (1536/64, 1024/32).
// ---------------------------------------------------------------------------
__global__ __launch_bounds__(256) void qkv_gemm(const float* __restrict__ x,
                                                const float* __restrict__ w,
                                                float* __restrict__ qb,
                                                float* __restrict__ kb,
                                                float* __restrict__ vb) {
    const int C = 512;
    const int tid  = threadIdx.x;
    const int wave = tid >> 5;
    const int lane = tid & 31;
    const int half = lane >> 4;   // 0: K=0,1  1: K=2,3
    const int l    = lane & 15;
    const int wm = wave >> 2, wn = wave & 3;
    const int m0 = blockIdx.y * 32 + wm * 16;
    const int n0 = blockIdx.x * 64 + wn * 16;

    const float* ap = x + (m0 + l) * C + 2 * half;   // A[m0+l][k + 2*half]
    const float* bp = w + (n0 + l) * C + 2 * half;   // B[k][n0+l] = w[(n0+l)*C + k]

    v8f acc = {};
#pragma unroll 4
    for (int k = 0; k < C; k += 4) {
        v2f a = *(const v2f*)(ap + k);
        v2f b = *(const v2f*)(bp + k);
        acc = WMMA_F32(a, b, acc);
    }

    // Scatter D tile: col c3 selects {q,k,v} and (head, dh); row m -> (b, n).
    const int c3 = n0 + l;
    const int kk = c3 >> 9;
    const int h  = (c3 >> 6) & 7;
    const int dh = c3 & 63;
    float* base = (kk == 0) ? qb : ((kk == 1) ? kb : vb);
#pragma unroll
    for (int r = 0; r < 8; ++r) {
        const int m  = m0 + r + 8 * half;
        const int b_ = m >> 8;
        const int n  = m & 255;
        base[(((b_ * 8 + h) * 256) + n) * 64 + dh] = acc[r];
    }
}

// ---------------------------------------------------------------------------
// Kernel 2: Fourier integral kernel scores + key-dim normalization.
// One block per (b,h,query). Thread j handles key j: 64 sinc terms, then
// wave32 shuffle reduction + LDS across 8 waves for the row sum.
// ---------------------------------------------------------------------------
__global__ __launch_bounds__(256) void fourier_scores(const float* __restrict__ qb,
                                                      const float* __restrict__ kb,
                                                      const float* __restrict__ pR,
                                                      float* __restrict__ attn) {
    __shared__ float qsh[64];
    __shared__ float wsum[8];

    const int tid = threadIdx.x;
    const int bh  = blockIdx.x >> 8;
    const int qi  = blockIdx.x & 255;
    const float R = pR[0];

    if (tid < 64) qsh[tid] = qb[(bh * 256 + qi) * 64 + tid];
    __syncthreads();

    const float4* kp = (const float4*)(kb + (size_t)(bh * 256 + tid) * 64);
    float p = 1.0f;
#pragma unroll
    for (int d4 = 0; d4 < 16; ++d4) {
        const float4 kv = kp[d4];
        float df;
        df = qsh[4 * d4 + 0] - kv.x;
        p *= (__builtin_fabsf(df) < 1e-8f) ? R : __fdividef(__sinf(R * df), df);
        df = qsh[4 * d4 + 1] - kv.y;
        p *= (__builtin_fabsf(df) < 1e-8f) ? R : __fdividef(__sinf(R * df), df);
        df = qsh[4 * d4 + 2] - kv.z;
        p *= (__builtin_fabsf(df) < 1e-8f) ? R : __fdividef(__sinf(R * df), df);
        df = qsh[4 * d4 + 3] - kv.w;
        p *= (__builtin_fabsf(df) < 1e-8f) ? R : __fdividef(__sinf(R * df), df);
    }

    float a4 = p * p;
    a4 *= a4;

    // wave32 reduction, then cross-wave via LDS
    float s = a4;
#pragma unroll
    for (int off = 16; off > 0; off >>= 1) s += __shfl_down(s, off, 32);
    if ((tid & 31) == 0) wsum[tid >> 5] = s;
    __syncthreads();
    float tot = 0.0f;
#pragma unroll
    for (int i = 0; i < 8; ++i) tot += wsum[i];

    attn[((size_t)(bh * 256 + qi)) * 256 + tid] = a4 / (tot + 1e-6f);
}

// ---------------------------------------------------------------------------
// Kernel 3: out = attn(256x256) @ v(256x64) per (b,h); result written directly
// into (B,N,C) layout for the projection GEMM. Block tile 32(M) x 64(N=Dh).
// Grid = (256/32, B*H).
// ---------------------------------------------------------------------------
__global__ __launch_bounds__(256) void attn_v_gemm(const float* __restrict__ attn,
                                                   const float* __restrict__ vb,
                                                   float* __restrict__ x2) {
    const int tid  = threadIdx.x;
    const int wave = tid >> 5;
    const int lane = tid & 31;
    const int half = lane >> 4;
    const int l    = lane & 15;
    const int wm = wave >> 2, wn = wave & 3;
    const int bh = blockIdx.y;
    const int m0 = blockIdx.x * 32 + wm * 16;
    const int n0 = wn * 16;

    const float* ap = attn + ((size_t)(bh * 256 + m0 + l)) * 256 + 2 * half;
    const float* bp = vb + (size_t)(bh * 256) * 64 + n0 + l;  // B[k][n] at bp + k*64

    v8f acc = {};
#pragma unroll 4
    for (int k = 0; k < 256; k += 4) {
        v2f a = *(const v2f*)(ap + k);
        const int kk = k + 2 * half;
        v2f b = { bp[kk * 64], bp[(kk + 1) * 64] };
        acc = WMMA_F32(a, b, acc);
    }

    const int b_ = bh >> 3;
    const int h  = bh & 7;
    const int d  = n0 + l;
#pragma unroll
    for (int r = 0; r < 8; ++r) {
        const int qi = m0 + r + 8 * half;
        x2[((size_t)(b_ * 256 + qi)) * 512 + h * 64 + d] = acc[r];
    }
}

// ---------------------------------------------------------------------------
// Kernel 4: final = x2(1024x512) @ w_proj^T(512x512) + b_proj.
// Grid = (512/64, 1024/32).
// ---------------------------------------------------------------------------
__global__ __launch_bounds__(256) void proj_gemm(const float* __restrict__ x2,
                                                 const float* __restrict__ w,
                                                 const float* __restrict__ bias,
                                                 float* __restrict__ out) {
    const int C = 512;
    const int tid  = threadIdx.x;
    const int wave = tid >> 5;
    const int lane = tid & 31;
    const int half = lane >> 4;
    const int l    = lane & 15;
    const int wm = wave >> 2, wn = wave & 3;
    const int m0 = blockIdx.y * 32 + wm * 16;
    const int n0 = blockIdx.x * 64 + wn * 16;

    const float* ap = x2 + (m0 + l) * C + 2 * half;
    const float* bp = w  + (n0 + l) * C + 2 * half;

    v8f acc = {};
#pragma unroll 4
    for (int k = 0; k < C; k += 4) {
        v2f a = *(const v2f*)(ap + k);
        v2f b = *(const v2f*)(bp + k);
        acc = WMMA_F32(a, b, acc);
    }

    const int c = n0 + l;
    const float bv = bias[c];
#pragma unroll
    for (int r = 0; r < 8; ++r) {
        const int m = m0 + r + 8 * half;
        out[(size_t)m * C + c] = acc[r] + bv;
    }
}

// ---------------------------------------------------------------------------
extern "C" void kernel_launch(void* const* d_in, const int* in_sizes, int n_in,
                              void* d_out, int out_size, void* d_ws, size_t ws_size,
                              hipStream_t stream) {
    (void)in_sizes; (void)n_in; (void)out_size; (void)ws_size;

    const float* x      = (const float*)d_in[0];  // (4,256,512)
    const float* w_qkv  = (const float*)d_in[1];  // (1536,512)
    const float* w_proj = (const float*)d_in[2];  // (512,512)
    const float* b_proj = (const float*)d_in[3];  // (512,)
    const float* paramR = (const float*)d_in[4];  // (1,)
    float* out = (float*)d_out;                   // (4,256,512)

    // Workspace layout (floats): q,k,v (B,H,N,Dh) + attn (B,H,N,N) + ctx (B,N,C)
    float* ws   = (float*)d_ws;
    float* qb   = ws;                 // 524288
    float* kb   = ws + 524288;        // 524288
    float* vb   = ws + 1048576;       // 524288
    float* attn = ws + 1572864;       // 2097152
    float* x2   = ws + 3670016;       // 524288   (total 16 MB)

    qkv_gemm<<<dim3(24, 32), 256, 0, stream>>>(x, w_qkv, qb, kb, vb);
    fourier_scores<<<dim3(4 * 8 * 256), 256, 0, stream>>>(qb, kb, paramR, attn);
    attn_v_gemm<<<dim3(8, 32), 256, 0, stream>>>(attn, vb, x2);
    proj_gemm<<<dim3(8, 32), 256, 0, stream>>>(x2, w_proj, b_proj, out);
}